// ShortConvolution_730144440447
// MI455X (gfx1250) — compile-verified
//
#include <hip/hip_runtime.h>

// Problem constants (match reference)
#define BB 4
#define TT 4096
#define DD 2048
#define WW 4

#define NCOL   (DD / 4)   // float4 columns per row = 512
#define BLOCK  256        // 8 wave32 per block
#define TCHUNK 64         // timesteps per block
#define PFDIST 8          // prefetch distance in rows (8 * 8KB = 64KB ahead)

#define LOG2E 1.4426950408889634f

// Native clang vector: guarantees b128 vector loads/stores (no scalarization)
typedef float v4f __attribute__((ext_vector_type(4)));

__device__ __forceinline__ float silu1(float v) {
    // v * sigmoid(v) = v / (1 + exp(-v)); exp via v_exp_f32, recip via v_rcp_f32
    float e = __builtin_amdgcn_exp2f(-v * LOG2E);
    return v * __builtin_amdgcn_rcpf(1.0f + e);
}

__global__ __launch_bounds__(BLOCK)
void ShortConvolution_kernel(const v4f* __restrict__ x4,   // [B, T, D/4]
                             const v4f* __restrict__ w4,   // [W, D/4]
                             v4f* __restrict__ y4,         // [B, T, D/4]
                             v4f* __restrict__ st4)        // [B, W, D/4]
{
    const int d4 = blockIdx.x * BLOCK + threadIdx.x;   // 0 .. NCOL-1
    const int b  = blockIdx.z;
    const int t0 = blockIdx.y * TCHUNK;

    // Depthwise taps for this float4 of channels (w[k] multiplies x[t+k-(W-1)])
    const v4f w0 = w4[0 * NCOL + d4];
    const v4f w1 = w4[1 * NCOL + d4];
    const v4f w2 = w4[2 * NCOL + d4];
    const v4f w3 = w4[3 * NCOL + d4];

    const v4f* xb = x4 + (size_t)b * TT * NCOL + d4;
    v4f*       yb = y4 + (size_t)b * TT * NCOL + d4;

    const v4f zero = {0.f, 0.f, 0.f, 0.f};
    // Sliding history: x[t-3], x[t-2], x[t-1] (zero before t=0)
    v4f xm3 = (t0 >= 3) ? xb[(size_t)(t0 - 3) * NCOL] : zero;
    v4f xm2 = (t0 >= 2) ? xb[(size_t)(t0 - 2) * NCOL] : zero;
    v4f xm1 = (t0 >= 1) ? xb[(size_t)(t0 - 1) * NCOL] : zero;

#pragma unroll 4
    for (int t = t0; t < t0 + TCHUNK; ++t) {
        // Streamed-once input: non-temporal load (TH_NT) — x+y working set
        // (268 MB) exceeds the 192 MB L2, so RT fills would only thrash.
        const v4f cur = __builtin_nontemporal_load(&xb[(size_t)t * NCOL]);

        // Keep the HBM stream deep: prefetch PFDIST rows ahead (clamped in-bounds)
        int tp = t + PFDIST;
        if (tp > TT - 1) tp = TT - 1;
        __builtin_prefetch((const void*)&xb[(size_t)tp * NCOL], 0, 3);

        // y[t] = w3*x[t] + w2*x[t-1] + w1*x[t-2] + w0*x[t-3]
        v4f acc = __builtin_elementwise_fma(
            cur, w3,
            __builtin_elementwise_fma(
                xm1, w2,
                __builtin_elementwise_fma(xm2, w1, xm3 * w0)));

        acc.x = silu1(acc.x);
        acc.y = silu1(acc.y);
        acc.z = silu1(acc.z);
        acc.w = silu1(acc.w);

        // Streamed-once output, never re-read: non-temporal store
        __builtin_nontemporal_store(acc, &yb[(size_t)t * NCOL]);

        xm3 = xm2;
        xm2 = xm1;
        xm1 = cur;
    }

    // Final-state emit hoisted out of the hot loop: only the 8 tail blocks
    // (blockIdx.y == last) re-read x[:, T-W:, :] (128 KB total) and store it.
    if (t0 + TCHUNK == TT) {
#pragma unroll
        for (int r = 0; r < WW; ++r) {
            const v4f v = xb[(size_t)(TT - WW + r) * NCOL];
            st4[((size_t)b * WW + (size_t)r) * NCOL + d4] = v;
        }
    }
}

extern "C" void kernel_launch(void* const* d_in, const int* in_sizes, int n_in,
                              void* d_out, int out_size, void* d_ws, size_t ws_size,
                              hipStream_t stream) {
    (void)in_sizes; (void)n_in; (void)out_size; (void)d_ws; (void)ws_size;

    const float* x = (const float*)d_in[0];   // [B, T, D] f32
    const float* w = (const float*)d_in[1];   // [W, D]    f32

    float* y  = (float*)d_out;                         // [B, T, D]
    float* st = (float*)d_out + (size_t)BB * TT * DD;  // [B, W, D]

    dim3 grid(NCOL / BLOCK, TT / TCHUNK, BB);  // (2, 64, 4) = 512 blocks
    dim3 block(BLOCK);

    ShortConvolution_kernel<<<grid, block, 0, stream>>>(
        (const v4f*)x, (const v4f*)w, (v4f*)y, (const_cast<v4f*>((v4f*)st)));
}